// MultiMPNN_17420387352966
// MI455X (gfx1250) — compile-verified
//
#include <hip/hip_runtime.h>
#include <hip/hip_bf16.h>
#include <math.h>

// Problem constants (from reference): B=4, N=512, Z=64, E=16, H=128
#define B_ 4
#define N_ 512
#define Z_ 64
#define E_ 16
#define H_ 128
#define ROWS_ (B_ * N_)   // 2048 flattened (b,n) rows

typedef float v2f __attribute__((ext_vector_type(2)));
typedef float v8f __attribute__((ext_vector_type(8)));

// D = A(16x4 f32) * B(4x16 f32) + C(16x16 f32), wave32 WMMA.
// A frag: lane l holds A[l%16][2*(l/16) + p] in VGPR p  -> contiguous float2 per lane
// B frag: lane l holds B[2*(l/16) + p][l%16] in VGPR p
// C/D:    lane l, VGPR r holds D[r + 8*(l/16)][l%16]
static __device__ __forceinline__ v8f wmma_f32(v2f a, v2f b, v8f c) {
    return __builtin_amdgcn_wmma_f32_16x16x4_f32(
        /*neg_a=*/false, a, /*neg_b=*/false, b,
        /*c_mod=*/(short)0, c, /*reuse_a=*/false, /*reuse_b=*/false);
}

// ---------------------------------------------------------------------------
// Kernel 1: P[row,h] = z[row,:] @ W_j ;  Q[row,h] = z[row,:] @ W_i + b_msg
// grid = ROWS_/16 blocks, 256 threads (8 waves), wave w owns h-tile [16w,16w+16)
// ---------------------------------------------------------------------------
__global__ void mpnn_proj_ij(const float* __restrict__ z,
                             const float* __restrict__ W_msg,
                             const float* __restrict__ b_msg,
                             float* __restrict__ P,
                             float* __restrict__ Q) {
    const int tid  = threadIdx.x;
    const int wave = tid >> 5;
    const int lane = tid & 31;
    const int lm   = lane & 15;
    const int lh   = lane >> 4;
    const int h0   = wave * 16;
    const int mrow = blockIdx.x * 16 + lm;

    v8f ci = {0.f,0.f,0.f,0.f,0.f,0.f,0.f,0.f};
    v8f cj = ci;
    const float* zrow = z + (size_t)mrow * Z_;

#pragma unroll
    for (int t = 0; t < Z_ / 4; ++t) {
        const int kb = 4 * t + 2 * lh;                    // K base for this lane half
        v2f a = *(const v2f*)(zrow + kb);                 // z[row, kb..kb+1]
        const float* wi = W_msg + (size_t)kb * H_ + h0 + lm;          // W_i rows
        const float* wj = W_msg + (size_t)(Z_ + kb) * H_ + h0 + lm;   // W_j rows
        v2f bi = { wi[0], wi[H_] };
        v2f bj = { wj[0], wj[H_] };
        ci = wmma_f32(a, bi, ci);
        cj = wmma_f32(a, bj, cj);
    }

    const float bm = b_msg[h0 + lm];
#pragma unroll
    for (int r = 0; r < 8; ++r) {
        const int orow = blockIdx.x * 16 + r + 8 * lh;
        Q[(size_t)orow * H_ + h0 + lm] = ci[r] + bm;      // zi + b_msg
        P[(size_t)orow * H_ + h0 + lm] = cj[r];           // zj
    }
}

// ---------------------------------------------------------------------------
// Kernel 2: per (b,i):
//   agg[b,i,h] = Q[b,i,h] + max_j ( maskedP[j,h] + (e_feat[b,i,j,:] @ W_e)[h] )
// where maskedP[j,h] = P[b,j,h] + (adj[b,i,j] > 0 ? 0 : -inf) is staged in LDS
// per block, so both the zj term AND the mask ride in the WMMA accumulator:
// hot loop = 8 ds_load (C init) + 4 global_load_b64 (A) + 4 v_wmma + 8 v_max.
// grid = B_*N_ blocks (one per (b,i)), 256 threads (8 waves, one h-tile each).
// ---------------------------------------------------------------------------
__global__ void mpnn_agg_kernel(const float* __restrict__ e_feat,
                                const int*   __restrict__ adj,
                                const float* __restrict__ W_msg,
                                const float* __restrict__ P,
                                const float* __restrict__ Q,
                                float* __restrict__ agg) {
    extern __shared__ float sP[];                   // N_*H_ floats (256 KB)

    const int bi   = blockIdx.x;                    // b*N + i
    const int b    = bi >> 9;
    const int tid  = threadIdx.x;
    const int wave = tid >> 5;
    const int lane = tid & 31;
    const int lm   = lane & 15;
    const int lh   = lane >> 4;
    const int h0   = wave * 16;

    // Stage P[b,:,:] with the per-j adjacency mask pre-added.
    {
        const float4* src  = (const float4*)(P + (size_t)b * N_ * H_);
        const int*    arow = adj + (size_t)bi * N_;
        float4* dst = (float4*)sP;
        for (int idx = tid; idx < (N_ * H_) / 4; idx += blockDim.x) {
            float4 v = src[idx];
            const int j   = idx >> 5;               // H_/4 = 32 float4 per j row
            const float m = (arow[j] > 0) ? 0.0f : -INFINITY;
            v.x += m; v.y += m; v.z += m; v.w += m;
            dst[idx] = v;
        }
    }
    __syncthreads();

    // W_e B-fragments (rows 2Z..2Z+15 of W_msg), held in registers all loop.
    v2f be[4];
#pragma unroll
    for (int t = 0; t < 4; ++t) {
        const int kb = 4 * t + 2 * lh;
        const float* we = W_msg + (size_t)(2 * Z_ + kb) * H_ + h0 + lm;
        be[t] = (v2f){ we[0], we[H_] };
    }

    const float* erow = e_feat + (size_t)bi * N_ * E_;   // e_feat[b,i,0,0]
    float acc = -INFINITY;

    for (int j0 = 0; j0 < N_; j0 += 16) {
        const float* abase = erow + (size_t)(j0 + lm) * E_ + 2 * lh;
        if (j0 + 16 < N_)                                 // prefetch next tile
            __builtin_prefetch(erow + (size_t)(j0 + 16 + lm) * E_, 0, 1);

        // Accumulator initialized with maskedP tile: C/D elem r is row j0+r+8*lh.
        v8f c;
#pragma unroll
        for (int r = 0; r < 8; ++r)
            c[r] = sP[(j0 + r + 8 * lh) * H_ + h0 + lm];

#pragma unroll
        for (int t = 0; t < 4; ++t) {                     // K=16 in 4 WMMA steps
            v2f a = *(const v2f*)(abase + 4 * t);
            c = wmma_f32(a, be[t], c);
        }

#pragma unroll
        for (int r = 0; r < 8; ++r)                       // masked max over j
            acc = fmaxf(acc, c[r]);
    }

    // Fold the two half-waves (same h, disjoint j sets).
    acc = fmaxf(acc, __shfl_xor(acc, 16, 32));
    if (lh == 0)
        agg[(size_t)bi * H_ + h0 + lm] = acc + Q[(size_t)bi * H_ + h0 + lm];
}

// ---------------------------------------------------------------------------
// Kernel 3: out[row,:] = z[row,:] @ Wu_z + agg[row,:] @ Wu_h + b_upd
// ---------------------------------------------------------------------------
__global__ void mpnn_out_kernel(const float* __restrict__ z,
                                const float* __restrict__ agg,
                                const float* __restrict__ W_upd,
                                const float* __restrict__ b_upd,
                                float* __restrict__ out) {
    const int tid  = threadIdx.x;
    const int wave = tid >> 5;
    const int lane = tid & 31;
    const int lm   = lane & 15;
    const int lh   = lane >> 4;
    const int h0   = wave * 16;
    const int mrow = blockIdx.x * 16 + lm;

    v8f c = {0.f,0.f,0.f,0.f,0.f,0.f,0.f,0.f};

    const float* zrow = z + (size_t)mrow * Z_;
#pragma unroll
    for (int t = 0; t < Z_ / 4; ++t) {                    // z @ Wu_z (K=64)
        const int kb = 4 * t + 2 * lh;
        v2f a = *(const v2f*)(zrow + kb);
        const float* w = W_upd + (size_t)kb * H_ + h0 + lm;
        v2f bb = { w[0], w[H_] };
        c = wmma_f32(a, bb, c);
    }

    const float* arow = agg + (size_t)mrow * H_;
#pragma unroll
    for (int t = 0; t < H_ / 4; ++t) {                    // agg @ Wu_h (K=128)
        const int kb = 4 * t + 2 * lh;
        v2f a = *(const v2f*)(arow + kb);
        const float* w = W_upd + (size_t)(Z_ + kb) * H_ + h0 + lm;
        v2f bb = { w[0], w[H_] };
        c = wmma_f32(a, bb, c);
    }

    const float bu = b_upd[h0 + lm];
#pragma unroll
    for (int r = 0; r < 8; ++r) {
        const int orow = blockIdx.x * 16 + r + 8 * lh;
        out[(size_t)orow * H_ + h0 + lm] = c[r] + bu;
    }
}

// ---------------------------------------------------------------------------
extern "C" void kernel_launch(void* const* d_in, const int* in_sizes, int n_in,
                              void* d_out, int out_size, void* d_ws, size_t ws_size,
                              hipStream_t stream) {
    const float* z      = (const float*)d_in[0];   // (B,N,Z)
    const float* e_feat = (const float*)d_in[1];   // (B,N,N,E)
    const int*   adj    = (const int*)  d_in[2];   // (B,N,N)
    const float* W_msg  = (const float*)d_in[3];   // (2Z+E, H)
    const float* b_msg  = (const float*)d_in[4];   // (H,)
    const float* W_upd  = (const float*)d_in[5];   // (Z+H, H)
    const float* b_upd  = (const float*)d_in[6];   // (H,)
    float* out = (float*)d_out;                    // (B,N,H)

    float* ws  = (float*)d_ws;
    float* P   = ws;                               // ROWS_*H_  (zj)
    float* Q   = ws + (size_t)ROWS_ * H_;          // ROWS_*H_  (zi + b_msg)
    float* agg = ws + 2 * (size_t)ROWS_ * H_;      // ROWS_*H_

    mpnn_proj_ij<<<ROWS_ / 16, 256, 0, stream>>>(z, W_msg, b_msg, P, Q);

    const size_t shmem = (size_t)N_ * H_ * sizeof(float);   // 256 KB < 320 KB WGP LDS
    mpnn_agg_kernel<<<B_ * N_, 256, shmem, stream>>>(e_feat, adj, W_msg, P, Q, agg);

    mpnn_out_kernel<<<ROWS_ / 16, 256, 0, stream>>>(z, agg, W_upd, b_upd, out);
}